// vdWModule_11699490915121
// MI455X (gfx1250) — compile-verified
//
#include <hip/hip_runtime.h>

typedef __attribute__((ext_vector_type(2))) float v2f;
typedef __attribute__((ext_vector_type(8))) float v8f;

#define N_NODES 2048
#define C_DIM   64

// ---------------------------------------------------------------------------
// Kernel 1: per-node projections  P = embs(2048x64) x F(64x16)
//   F columns: 0 = radius_factor[:C] (lo), 1 = radius_factor[C:] (hi),
//              2 = depth_factor[:C]  (lo), 3 = depth_factor[C:]  (hi),
//              4..15 = zero (masked multiplicatively, no branches).
//   One wave computes a 16x16 tile via 16 fully-unrolled steps of
//   V_WMMA_F32_16X16X4_F32 (full f32 precision, K=64 accumulated).
// ---------------------------------------------------------------------------
__global__ __launch_bounds__(128) void node_proj_wmma(
    const float* __restrict__ embs,
    const float* __restrict__ depth_factor,
    const float* __restrict__ radius_factor,
    float* __restrict__ rl, float* __restrict__ rh,
    float* __restrict__ dl, float* __restrict__ dh)
{
  const int lane = threadIdx.x & 31;
  const int wave = (blockIdx.x * blockDim.x + threadIdx.x) >> 5;
  const int row0 = wave * 16;              // 16 nodes per wave
  const int m    = lane & 15;              // row (A) / col (B,C,D) within tile
  const int koff = (lane >> 4) << 1;       // lanes 16-31 hold K+2,K+3

  // Branch-free B setup: cols 0,1 from radius_factor, cols 2,3 from
  // depth_factor, odd cols use the upper half [C:2C]; cols >=4 are zeroed
  // by mval. All loads stay in bounds (off + k <= 127 < 128).
  const float* __restrict__ srcp = (m < 2) ? radius_factor : depth_factor;
  const int   off  = (m & 1) * C_DIM;
  const float mval = (m < 4) ? 1.0f : 0.0f;

  const float* __restrict__ arow = embs + (size_t)(row0 + m) * C_DIM;

  v8f acc = {};
#pragma unroll
  for (int kk = 0; kk < C_DIM; kk += 4) {
    const int k = kk + koff;               // even -> 8B aligned v2f loads
    v2f a = *(const v2f*)(arow + k);
    v2f f = *(const v2f*)(srcp + off + k);
    v2f b; b.x = mval * f.x; b.y = mval * f.y;
    // 8 args: (neg_a, A, neg_b, B, c_mod, C, reuse_a, reuse_b)
    acc = __builtin_amdgcn_wmma_f32_16x16x4_f32(
        false, a, false, b, (short)0, acc, false, false);
  }

  // D layout: VGPR v -> M = v + 8*(lane>=16), N = lane&15; keep cols 0..3.
  if (m < 4) {
    float* dst = (m == 0) ? rl : (m == 1) ? rh : (m == 2) ? dl : dh;
    const int rbase = row0 + ((lane >> 4) << 3);
#pragma unroll
    for (int v = 0; v < 8; ++v) dst[rbase + v] = acc[v];
  }
}

// ---------------------------------------------------------------------------
// Kernel 2: one block per row i; stream 2048 j's; deterministic LDS reduction.
// ---------------------------------------------------------------------------
__global__ __launch_bounds__(256) void pair_energy(
    const float* __restrict__ X,  const float* __restrict__ s0,
    const float* __restrict__ w0, const float* __restrict__ mask,
    const float* __restrict__ rl, const float* __restrict__ rh,
    const float* __restrict__ dl, const float* __restrict__ dh,
    float* __restrict__ partials)
{
  const int i = blockIdx.x;
  const float xi = X[i * 3 + 0], yi = X[i * 3 + 1], zi = X[i * 3 + 2];
  const float s0i = s0[i], w0i = w0[i];
  const float rhi = rh[i], dhi = dh[i];

  float sum = 0.0f;
  for (int j = threadIdx.x; j < N_NODES; j += 256) {
    const float dx = X[j * 3 + 0] - xi;
    const float dy = X[j * 3 + 1] - yi;
    const float dz = X[j * 3 + 2] - zi;
    // torch/jax add 1e-6 per coordinate before the sum -> +3e-6
    const float D = __builtin_sqrtf(dx * dx + dy * dy + dz * dz + 3e-6f);
    const float mk = mask[(size_t)i * N_NODES + j];

    const float sig_r = 1.0f / (1.0f + __expf(-(rhi + rl[j])));
    const float sig_d = 1.0f / (1.0f + __expf(-(dhi + dl[j])));

    const float s    = (s0i + s0[j]) * (0.8f * sig_r + 0.4f);
    const float repl = 5.0f * __expf(-0.3f * D * D * D) * mk;

    const float Dm = D - s;
    const float t  = Dm - 0.3f;
    const float attr = (__expf(-t * t) + __expf(-3.0f * Dm * Dm) +
                        __expf(-10.0f * Dm * Dm)) * (1.0f / 3.0f) * mk;

    const float w = __builtin_sqrtf(w0i * w0[j] + 1e-6f) * (sig_d + 0.5f);
    sum += repl - w * attr;   // = -w*attr + repl
  }

  __shared__ float red[256];
  red[threadIdx.x] = sum;
  __syncthreads();
  for (int sft = 128; sft > 0; sft >>= 1) {
    if (threadIdx.x < sft) red[threadIdx.x] += red[threadIdx.x + sft];
    __syncthreads();
  }
  if (threadIdx.x == 0) partials[i] = red[0];
}

// ---------------------------------------------------------------------------
// Kernel 3: fixed-order final reduction + NaN guard.
// ---------------------------------------------------------------------------
__global__ __launch_bounds__(256) void final_reduce(
    const float* __restrict__ partials, float* __restrict__ out)
{
  float sum = 0.0f;
  for (int k = threadIdx.x; k < N_NODES; k += 256) sum += partials[k];
  __shared__ float red[256];
  red[threadIdx.x] = sum;
  __syncthreads();
  for (int sft = 128; sft > 0; sft >>= 1) {
    if (threadIdx.x < sft) red[threadIdx.x] += red[threadIdx.x + sft];
    __syncthreads();
  }
  if (threadIdx.x == 0) {
    const float e = red[0];
    out[0] = (e != e) ? 1e-6f : e;   // isnan -> 1e-6
  }
}

// ---------------------------------------------------------------------------
// Host launcher.
// Input order: X, embs, w0, s0, paired_mask, depth_factor, radius_factor
// ---------------------------------------------------------------------------
extern "C" void kernel_launch(void* const* d_in, const int* in_sizes, int n_in,
                              void* d_out, int out_size, void* d_ws, size_t ws_size,
                              hipStream_t stream) {
  (void)in_sizes; (void)n_in; (void)out_size; (void)ws_size;
  const float* X             = (const float*)d_in[0];
  const float* embs          = (const float*)d_in[1];
  const float* w0            = (const float*)d_in[2];
  const float* s0            = (const float*)d_in[3];
  const float* paired_mask   = (const float*)d_in[4];
  const float* depth_factor  = (const float*)d_in[5];
  const float* radius_factor = (const float*)d_in[6];

  float* ws       = (float*)d_ws;
  float* rl       = ws + 0 * N_NODES;
  float* rh       = ws + 1 * N_NODES;
  float* dl       = ws + 2 * N_NODES;
  float* dh       = ws + 3 * N_NODES;
  float* partials = ws + 4 * N_NODES;

  // 2048 rows / 16 rows-per-wave / 4 waves-per-block = 32 blocks
  node_proj_wmma<<<32, 128, 0, stream>>>(embs, depth_factor, radius_factor,
                                         rl, rh, dl, dh);
  pair_energy<<<N_NODES, 256, 0, stream>>>(X, s0, w0, paired_mask,
                                           rl, rh, dl, dh, partials);
  final_reduce<<<1, 256, 0, stream>>>(partials, (float*)d_out);
}